// AttentionMechanism_37675453121101
// MI455X (gfx1250) — compile-verified
//
#include <hip/hip_runtime.h>
#include <hip/hip_bf16.h>

#define B_SZ   2048
#define T_SZ   32
#define TAU_SZ 1024
#define MSG_SZ 512
#define HID_SZ 1024
#define VD_SZ  128

typedef __attribute__((ext_vector_type(2))) float v2f;
typedef __attribute__((ext_vector_type(8))) float v8f;
typedef __attribute__((ext_vector_type(4))) int   v4i;

#ifndef __has_builtin
#define __has_builtin(x) 0
#endif
#if __has_builtin(__builtin_amdgcn_global_load_async_to_lds_b128)
#define HAVE_ASYNC_LDS 1
#else
#define HAVE_ASYNC_LDS 0
#endif

#define AS1 __attribute__((address_space(1)))
#define AS3 __attribute__((address_space(3)))

// ---------------------------------------------------------------------------
// WMMA f32 GEMM: C[M,N] = A[M,K] * op(B) + bias[N]
//   BT=true : B is [N,K] row-major (NT gemm, torch nn.Linear weight layout)
//   BT=false: B is [K,N] row-major (NN gemm)
// One wave per 16x16 tile, K-step 4 via V_WMMA_F32_16X16X4_F32.
// ---------------------------------------------------------------------------
template <bool BT>
__launch_bounds__(32)
__global__ void wmma_gemm_f32(const float* __restrict__ A,
                              const float* __restrict__ B,
                              const float* __restrict__ bias,
                              float* __restrict__ C,
                              int M, int N, int K) {
  const int lane = threadIdx.x & 31;
  const int half = lane >> 4;   // 0 or 1
  const int lm   = lane & 15;
  const int row0 = blockIdx.y * 16;
  const int col0 = blockIdx.x * 16;

  const float* ap = A + (size_t)(row0 + lm) * K + 2 * half;
  v8f acc = {};

  if (BT) {
    const float* bp = B + (size_t)(col0 + lm) * K + 2 * half;
    for (int k = 0; k < K; k += 4) {
      v2f a = *(const v2f*)(ap + k);
      v2f b = *(const v2f*)(bp + k);
      acc = __builtin_amdgcn_wmma_f32_16x16x4_f32(
          false, a, false, b, (short)0, acc, false, false);
    }
  } else {
    const float* bp = B + (size_t)(2 * half) * N + col0 + lm;
    for (int k = 0; k < K; k += 4) {
      v2f a = *(const v2f*)(ap + k);
      v2f b;
      b.x = bp[(size_t)k * N];
      b.y = bp[(size_t)(k + 1) * N];
      acc = __builtin_amdgcn_wmma_f32_16x16x4_f32(
          false, a, false, b, (short)0, acc, false, false);
    }
  }

  const float bz = bias ? bias[col0 + lm] : 0.0f;
  float* cp = C + (size_t)(row0 + 8 * half) * N + col0 + lm;
#pragma unroll
  for (int i = 0; i < 8; i++) cp[(size_t)i * N] = acc[i] + bz;
}

// ---------------------------------------------------------------------------
// Fused scores + online softmax + weighted tau accumulation.
// One 256-thread block (8 wave32) per batch row; tau[b] streamed once.
// Rows double-buffered in LDS via GLOBAL_LOAD_ASYNC_TO_LDS_B128 (ASYNCcnt),
// drained with s_wait_asynccnt before the buffer-publish barrier.
// wtau[b,c] = sum_t softmax_t(tau[b,t,:].qk[b,:]/32) * tau[b,t,c]
// (the q.bk term is constant over t and cancels in the softmax).
// ---------------------------------------------------------------------------
__launch_bounds__(256)
__global__ void attn_online_softmax(const float* __restrict__ tau,
                                    const float* __restrict__ qk,
                                    float* __restrict__ wtau) {
  const int b    = blockIdx.x;
  const int tid  = threadIdx.x;     // 0..255
  const int lane = tid & 31;
  const int wid  = tid >> 5;        // 0..7

  __shared__ float rowbuf[2][TAU_SZ];
  __shared__ float red[8];

  const float* taub = tau + (size_t)b * (T_SZ * TAU_SZ);

  float qkr[4], wacc[4] = {0.f, 0.f, 0.f, 0.f};
#pragma unroll
  for (int j = 0; j < 4; j++)
    qkr[j] = qk[(size_t)b * TAU_SZ + tid + 256 * j];

  float m = -__builtin_huge_valf();
  float d = 0.0f;

#if HAVE_ASYNC_LDS
  // each thread owns a 16B chunk of the 4 KiB row: bytes [tid*16, tid*16+16)
  __builtin_amdgcn_global_load_async_to_lds_b128(
      (AS1 v4i*)(taub + (size_t)tid * 4),
      (AS3 v4i*)(&rowbuf[0][tid * 4]), 0, 0);
  asm volatile("s_wait_asynccnt 0x0" ::: "memory");
#else
#pragma unroll
  for (int j = 0; j < 4; j++)
    rowbuf[0][tid + 256 * j] = taub[tid + 256 * j];
#endif
  __syncthreads();

  for (int t = 0; t < T_SZ; t++) {
    const int cur = t & 1;

    // kick off next row's memory->LDS transfer; overlaps entire compute phase
#if HAVE_ASYNC_LDS
    if (t + 1 < T_SZ)
      __builtin_amdgcn_global_load_async_to_lds_b128(
          (AS1 v4i*)(taub + (size_t)(t + 1) * TAU_SZ + tid * 4),
          (AS3 v4i*)(&rowbuf[cur ^ 1][tid * 4]), 0, 0);
#else
    if (t + 1 < T_SZ) {
#pragma unroll
      for (int j = 0; j < 4; j++)
        rowbuf[cur ^ 1][tid + 256 * j] =
            taub[(size_t)(t + 1) * TAU_SZ + tid + 256 * j];
    }
    if (t + 2 < T_SZ)  // gfx1250 global_prefetch_b8 two rows ahead
      __builtin_prefetch(taub + (size_t)(t + 2) * TAU_SZ + tid * 4, 0, 0);
#endif

    // score_t = tau[t,:].qk / 32
    float part = 0.0f;
#pragma unroll
    for (int j = 0; j < 4; j++)
      part += rowbuf[cur][tid + 256 * j] * qkr[j];
#pragma unroll
    for (int off = 16; off > 0; off >>= 1)
      part += __shfl_xor(part, off, 32);
    if (lane == 0) red[wid] = part;
    __syncthreads();

    float s = 0.0f;
#pragma unroll
    for (int w = 0; w < 8; w++) s += red[w];
    s *= 0.03125f;  // 1/sqrt(HID)

    // online softmax update (all threads compute identically)
    const float mn = fmaxf(m, s);
    const float sc = __expf(m - mn);
    const float e  = __expf(s - mn);
    d = d * sc + e;
#pragma unroll
    for (int j = 0; j < 4; j++)
      wacc[j] = wacc[j] * sc + e * rowbuf[cur][tid + 256 * j];
    m = mn;

#if HAVE_ASYNC_LDS
    // drain this wave's outstanding async before publishing the buffer
    asm volatile("s_wait_asynccnt 0x0" ::: "memory");
#endif
    __syncthreads();  // protects rowbuf + red[] for next iteration
  }

  const float inv = 1.0f / d;
#pragma unroll
  for (int j = 0; j < 4; j++)
    wtau[(size_t)b * TAU_SZ + tid + 256 * j] = wacc[j] * inv;
}

extern "C" void kernel_launch(void* const* d_in, const int* in_sizes, int n_in,
                              void* d_out, int out_size, void* d_ws, size_t ws_size,
                              hipStream_t stream) {
  const float* tau  = (const float*)d_in[0];  // [B, T*TAU]
  const float* msgs = (const float*)d_in[1];  // [B, MSG]
  const float* Wq   = (const float*)d_in[2];  // [HID, MSG]
  const float* bq   = (const float*)d_in[3];  // [HID]
  const float* Wk   = (const float*)d_in[4];  // [HID, TAU]
  // d_in[5] = bk: constant over t inside the softmax -> cancels exactly
  const float* Wv   = (const float*)d_in[6];  // [VDIM, TAU]
  const float* bv   = (const float*)d_in[7];  // [VDIM]
  float* out = (float*)d_out;                 // [B, VDIM]

  float* q    = (float*)d_ws;                 // [B, HID]   8 MiB
  float* qkp  = q   + (size_t)B_SZ * HID_SZ;  // [B, TAU]   8 MiB
  float* wtau = qkp + (size_t)B_SZ * TAU_SZ;  // [B, TAU]   8 MiB

  // 1) q = msgs @ Wq^T + bq                (NT, M=2048 N=1024 K=512)
  wmma_gemm_f32<true><<<dim3(HID_SZ / 16, B_SZ / 16), 32, 0, stream>>>(
      msgs, Wq, bq, q, B_SZ, HID_SZ, MSG_SZ);

  // 2) qk = q @ Wk                         (NN, M=2048 N=1024 K=1024)
  wmma_gemm_f32<false><<<dim3(TAU_SZ / 16, B_SZ / 16), 32, 0, stream>>>(
      q, Wk, nullptr, qkp, B_SZ, TAU_SZ, HID_SZ);

  // 3) fused scores/softmax/weighted-sum over tau (single 256 MiB stream)
  attn_online_softmax<<<B_SZ, 256, 0, stream>>>(tau, qkp, wtau);

  // 4) out = wtau @ Wv^T + bv              (NT, M=2048 N=128 K=1024)
  wmma_gemm_f32<true><<<dim3(VD_SZ / 16, B_SZ / 16), 32, 0, stream>>>(
      wtau, Wv, bv, out, B_SZ, VD_SZ, TAU_SZ);
}